// Network_47227460387324
// MI455X (gfx1250) — compile-verified
//
#include <hip/hip_runtime.h>
#include <hip/hip_bf16.h>

#define N_USERS 100000
#define N_ITEMS 50000
#define N_NODES (N_USERS + N_ITEMS)
#define D 32
#define N_EDGES 8000000
#define BATCH 65536
#define NELEM (N_NODES * D)   // 4,800,000 floats per node matrix

typedef __attribute__((ext_vector_type(2))) float v2f;
typedef __attribute__((ext_vector_type(8))) float v8f;

// ---------------------------------------------------------------------------
// 1) Concatenate [user_emb; item_emb] into x and into the running sum.
// ---------------------------------------------------------------------------
__global__ __launch_bounds__(256)
void init_concat(const float* __restrict__ ue, const float* __restrict__ ie,
                 float* __restrict__ x, float* __restrict__ s) {
    int i = blockIdx.x * 256 + threadIdx.x;
    if (i >= NELEM) return;
    int n = i >> 5;  // D == 32
    float v = (n < N_USERS) ? ue[i] : ie[i - N_USERS * D];
    x[i] = v;
    s[i] = v;
}

__global__ __launch_bounds__(256)
void zero_buf(float* __restrict__ p) {
    int i = blockIdx.x * 256 + threadIdx.x;
    if (i < NELEM) p[i] = 0.0f;
}

// ---------------------------------------------------------------------------
// 2) SpMM scatter: y[rows[e], :] += vals[e] * x[cols[e], :]
//    One wave32 per edge, lane == embedding dim.
//    x/y are L2-resident (19.2 MB each); atomics resolve in L2.
// ---------------------------------------------------------------------------
__global__ __launch_bounds__(256)
void spmm_scatter(const int* __restrict__ rows, const int* __restrict__ cols,
                  const float* __restrict__ vals,
                  const float* __restrict__ x, float* __restrict__ y) {
    int tid = blockIdx.x * 256 + threadIdx.x;
    int e = tid >> 5;        // edge index (8 edges per 256-thread block)
    int d = tid & 31;        // dim
    if (e >= N_EDGES) return;
    int r = rows[e];
    int c = cols[e];
    float v = vals[e];
    float contrib = v * x[(size_t)c * D + d];
    unsafeAtomicAdd(&y[(size_t)r * D + d], contrib);  // -> global_atomic_add_f32
}

__global__ __launch_bounds__(256)
void accum_sum(float* __restrict__ s, const float* __restrict__ y) {
    int i = blockIdx.x * 256 + threadIdx.x;
    if (i < NELEM) s[i] += y[i];
}

// ---------------------------------------------------------------------------
// 3) Batched dot products on the matrix pipe.
//    One wave per 16 outputs: C(16x16) = U(16x32) * V^T(32x16) via 8x
//    V_WMMA_F32_16X16X4_F32, take the diagonal, scale by 1/16 (the /4
//    LightGCN normalization folded for both operands), then affine.
//
//    A (16x4 f32) lane layout: lanes 0-15 hold {K0,K1} of row M=lane,
//    lanes 16-31 hold {K2,K3} of row M=lane-16. B (4x16) symmetric with
//    N = lane%16.  C/D: lanes 0-15: VGPR v = (M=v, N=lane);
//    lanes 16-31: VGPR v = (M=v+8, N=lane-16).
// ---------------------------------------------------------------------------
__global__ __launch_bounds__(256)
void gather_dot_wmma(const float* __restrict__ s,
                     const int* __restrict__ users, const int* __restrict__ items,
                     const float* __restrict__ stds, const float* __restrict__ means,
                     float* __restrict__ out) {
    int lane = threadIdx.x & 31;
    int g = blockIdx.x * 8 + (threadIdx.x >> 5);   // group of 16 outputs
    int m = (lane < 16) ? lane : (lane - 16);       // row within tile
    int b = g * 16 + m;
    int urow = users[b];
    int irow = N_USERS + items[b];
    const float* up = s + (size_t)urow * D;
    const float* ip = s + (size_t)irow * D;
    int kofs = (lane < 16) ? 0 : 2;

    v8f c = {};
#pragma unroll
    for (int k = 0; k < 8; ++k) {
        v2f a, bb;
        a.x  = up[4 * k + kofs];
        a.y  = up[4 * k + kofs + 1];
        bb.x = ip[4 * k + kofs];
        bb.y = ip[4 * k + kofs + 1];
        c = __builtin_amdgcn_wmma_f32_16x16x4_f32(
                /*neg_a=*/false, a, /*neg_b=*/false, bb,
                /*c_mod=*/(short)0, c, /*reuse_a=*/false, /*reuse_b=*/false);
    }

    // Diagonal extraction: diag i<8 at (lane=i, VGPR=i); diag i>=8 at
    // (lane=i+16, VGPR=i-8).
    bool active = (lane < 8) || (lane >= 24);
    int j = (lane < 8) ? lane : (lane - 24);   // VGPR index within c
    int i = (lane < 8) ? lane : (lane - 16);   // diagonal index
    float diag;
    switch (j) {
        case 0: diag = c[0]; break;
        case 1: diag = c[1]; break;
        case 2: diag = c[2]; break;
        case 3: diag = c[3]; break;
        case 4: diag = c[4]; break;
        case 5: diag = c[5]; break;
        case 6: diag = c[6]; break;
        default: diag = c[7]; break;
    }
    if (active) {
        int ob = g * 16 + i;
        int u = users[ob];
        out[ob] = diag * 0.0625f * stds[u] + means[u];
    }
}

// ---------------------------------------------------------------------------
// Host-side orchestration (graph-capture safe: kernels only, all on `stream`).
// ---------------------------------------------------------------------------
extern "C" void kernel_launch(void* const* d_in, const int* in_sizes, int n_in,
                              void* d_out, int out_size, void* d_ws, size_t ws_size,
                              hipStream_t stream) {
    const int*   users    = (const int*)d_in[0];
    const int*   items    = (const int*)d_in[1];
    const int*   rows     = (const int*)d_in[2];
    const int*   cols     = (const int*)d_in[3];
    const float* vals     = (const float*)d_in[4];
    const float* user_emb = (const float*)d_in[5];
    const float* item_emb = (const float*)d_in[6];
    const float* means    = (const float*)d_in[7];
    const float* stds     = (const float*)d_in[8];
    float*       out      = (float*)d_out;

    float* bufA = (float*)d_ws;
    float* bufB = bufA + (size_t)NELEM;
    float* sum  = bufB + (size_t)NELEM;   // 3 * 19.2 MB = 57.6 MB of workspace

    const int egrid    = (NELEM + 255) / 256;      // elementwise grids
    const int spmmgrid = N_EDGES / 8;              // 8 edges per 256-thread block

    init_concat<<<egrid, 256, 0, stream>>>(user_emb, item_emb, bufA, sum);

    float* x = bufA;
    float* y = bufB;
    for (int layer = 0; layer < 3; ++layer) {
        zero_buf<<<egrid, 256, 0, stream>>>(y);
        spmm_scatter<<<spmmgrid, 256, 0, stream>>>(rows, cols, vals, x, y);
        accum_sum<<<egrid, 256, 0, stream>>>(sum, y);
        float* t = x; x = y; y = t;
    }

    // B/16 = 4096 wave-groups, 8 per 256-thread block -> 512 blocks.
    gather_dot_wmma<<<BATCH / 16 / 8, 256, 0, stream>>>(sum, users, items,
                                                        stds, means, out);
}